// Decoder_39668317945935
// MI455X (gfx1250) — compile-verified
//
#include <hip/hip_runtime.h>
#include <cstdint>
#include <cstddef>

// ---------------- problem constants ----------------
#define B_IMG   8
#define N_ANC   120000
#define N_CLS   80
#define K_TOP   1000
#define BN      (B_IMG * N_ANC)          // 960000
#define HBINS   2048
#define HLO     (-10.0f)
#define HSCALE  (HBINS / 20.0f)          // bins over [-10, 10] in logit space
#define CANDMAX 4096
#define LOGIT_THRES (-2.9444389792f)     // log(0.05/0.95)  (unused directly; kept for doc)

// TDM descriptor vector types (probe-verified builtin signatures)
typedef unsigned int u32x4 __attribute__((ext_vector_type(4)));
typedef int          i32x8 __attribute__((ext_vector_type(8)));
typedef int          i32x4 __attribute__((ext_vector_type(4)));

// Padded-row LDS geometry for the TDM tile:
//   tile row = 80 dwords, pad 4 dwords every 16 stored dwords -> 100 dwords/row.
//   gcd(100,64)=4 -> 2-way LDS bank conflict instead of 8-way at raw stride 80.
#define ROWS_PER_BLK 128
#define ROW_DW       100     // 80 + 5*4 pad dwords

// ---------------- kernel 0: zero counters/histograms ----------------
__global__ void zero_meta(int* hist, int* cutoff, int* cand_count) {
    int g = blockIdx.x * 256 + threadIdx.x;
    if (g < B_IMG * HBINS) hist[g] = 0;
    if (g < B_IMG) { cutoff[g] = 0; cand_count[g] = 0; }
}

// ---------------- kernel 1: TDM-staged class reduce + box decode ----------------
// One 128-thread block handles 128 anchors. A single tensor_load_to_lds DMA
// streams the 128x80 f32 logit tile into LDS (padded); threads then reduce
// max/argmax over 80 classes from LDS and decode boxes from global float4s.
__global__ __launch_bounds__(ROWS_PER_BLK)
void score_decode(const float* __restrict__ cls_logits,
                  const float4* __restrict__ reg_preds,
                  const float4* __restrict__ anchors,
                  float* __restrict__ scores_ws,   // max logit per anchor
                  int* __restrict__ cls_ws,        // argmax + 1
                  float4* __restrict__ boxes_ws) {
    __shared__ __align__(16) float lds[ROWS_PER_BLK * ROW_DW];   // 51200 B

    const int t    = threadIdx.x;
    const int row0 = blockIdx.x * ROWS_PER_BLK;

    if (t < 32) {   // wave 0 issues the DMA (TDM ignores EXEC; one issue per wave)
        unsigned long long gaddr =
            (unsigned long long)(uintptr_t)(cls_logits + (size_t)row0 * N_CLS);
        unsigned lds_off  = (unsigned)(uintptr_t)&lds[0];   // flat addr low 32 = LDS offset
        unsigned rows_rem = (unsigned)(BN - row0);          // OOB rows read as zero

        // D# group 0: count=1 | lds_addr | global_addr[56:0] | type=2
        u32x4 g0 = { 1u,
                     lds_off,
                     (unsigned)(gaddr & 0xffffffffu),
                     (unsigned)((gaddr >> 32) & 0x1ffffffu) | (2u << 30) };
        // D# group 1: data_size=4B(2), pad_enable, pad_interval=16dw(3), pad_amount=4dw(3),
        //             tensor_dim0=80, tensor_dim1=rows_rem, tile=80x128, dim0_stride=80
        i32x8 g1 = { (int)((2u << 16) | (1u << 20) | (3u << 22) | (3u << 25)),
                     (int)((80u & 0xffffu) << 16),
                     (int)((rows_rem & 0xffffu) << 16),
                     (int)(((rows_rem >> 16) & 0xffffu) | (80u << 16)),
                     (int)ROWS_PER_BLK,      // tile_dim1=128, tile_dim2=0
                     80,                     // tensor_dim0_stride lo
                     0, 0 };
        i32x4 gz = { 0, 0, 0, 0 };           // groups 2/3 unused (2-D tile)
#if defined(__clang_major__) && (__clang_major__ >= 23)
        i32x8 gz8 = { 0, 0, 0, 0, 0, 0, 0, 0 };
        __builtin_amdgcn_tensor_load_to_lds(g0, g1, gz, gz, gz8, 0);
#else
        __builtin_amdgcn_tensor_load_to_lds(g0, g1, gz, gz, 0);
#endif
        __builtin_amdgcn_s_wait_tensorcnt(0);
    }
    __syncthreads();

    // max/argmax over 80 classes (monotone sigmoid deferred to the end)
    float best = -3.0e38f;
    int   bi   = 0;
    const int base = t * ROW_DW;
#pragma unroll
    for (int k = 0; k < 20; ++k) {
        const int dw = base + 4 * k + (k >> 2) * 4;          // 16B-aligned
        float4 v = *reinterpret_cast<const float4*>(&lds[dw]);
        const int c = 4 * k;
        if (v.x > best) { best = v.x; bi = c + 0; }
        if (v.y > best) { best = v.y; bi = c + 1; }
        if (v.z > best) { best = v.z; bi = c + 2; }
        if (v.w > best) { best = v.w; bi = c + 3; }
    }

    const int g = row0 + t;
    const int b = g / N_ANC;
    const int n = g - b * N_ANC;

    // decode: std = [0.1,0.1,0.2,0.2]
    float4 a = anchors[n];
    float aw  = a.z - a.x, ah = a.w - a.y;
    float acx = a.x + 0.5f * aw, acy = a.y + 0.5f * ah;
    float4 r = reg_preds[g];
    float cx = r.x * 0.1f * aw + acx;
    float cy = r.y * 0.1f * ah + acy;
    float w  = __expf(r.z * 0.2f) * aw;
    float h  = __expf(r.w * 0.2f) * ah;

    boxes_ws[g]  = make_float4(cx - 0.5f * w, cy - 0.5f * h, cx + 0.5f * w, cy + 0.5f * h);
    scores_ws[g] = best;
    cls_ws[g]    = bi + 1;
}

// ---------------- kernel 2: per-image logit histogram ----------------
__global__ void hist_kernel(const float* __restrict__ scores_ws, int* __restrict__ hist) {
    int g = blockIdx.x * 256 + threadIdx.x;
    if (g >= BN) return;
    int b = g / N_ANC;
    float s = scores_ws[g];
    int bin = (int)((s - HLO) * HSCALE);
    bin = min(HBINS - 1, max(0, bin));
    atomicAdd(&hist[b * HBINS + bin], 1);
}

// ---------------- kernel 3: find cutoff bin (>= K items at/above it) ----------------
__global__ void cutoff_kernel(const int* __restrict__ hist, int* __restrict__ cutoff) {
    if (threadIdx.x != 0) return;
    int b = blockIdx.x;
    int acc = 0, cut = 0;
    for (int bin = HBINS - 1; bin >= 0; --bin) {
        acc += hist[b * HBINS + bin];
        if (acc >= K_TOP) { cut = bin; break; }
    }
    cutoff[b] = cut;
}

// ---------------- kernel 4: compact candidates at/above cutoff ----------------
__global__ void compact_kernel(const float* __restrict__ scores_ws,
                               const int* __restrict__ cutoff,
                               int* __restrict__ cand_count,
                               float* __restrict__ cand_s,
                               int* __restrict__ cand_i) {
    int g = blockIdx.x * 256 + threadIdx.x;
    if (g >= BN) return;
    int b = g / N_ANC;
    float s = scores_ws[g];
    int bin = (int)((s - HLO) * HSCALE);
    bin = min(HBINS - 1, max(0, bin));
    if (bin >= cutoff[b]) {
        int pos = atomicAdd(&cand_count[b], 1);
        if (pos < CANDMAX) {
            cand_s[b * CANDMAX + pos] = s;
            cand_i[b * CANDMAX + pos] = g - b * N_ANC;
        }
    }
}

// ---------------- kernel 5: bitonic sort candidates, emit sorted top-K ----------------
__global__ __launch_bounds__(1024)
void topk_kernel(const float* __restrict__ cand_s, const int* __restrict__ cand_i,
                 const int* __restrict__ cand_count,
                 const int* __restrict__ cls_ws, const float4* __restrict__ boxes_ws,
                 float* __restrict__ top_s, int* __restrict__ top_c,
                 float4* __restrict__ top_b) {
    __shared__ float ss[CANDMAX];
    __shared__ int   ii[CANDMAX];
    const int b = blockIdx.x, tid = threadIdx.x;
    int M = cand_count[b];
    if (M > CANDMAX) M = CANDMAX;

    for (int t = tid; t < CANDMAX; t += 1024) {
        if (t < M) { ss[t] = cand_s[b * CANDMAX + t]; ii[t] = cand_i[b * CANDMAX + t]; }
        else       { ss[t] = -3.0e38f;                ii[t] = -1; }
    }
    __syncthreads();

    // bitonic sort, descending
    for (int k = 2; k <= CANDMAX; k <<= 1) {
        for (int j = k >> 1; j > 0; j >>= 1) {
            for (int t = tid; t < CANDMAX; t += 1024) {
                int x = t ^ j;
                if (x > t) {
                    bool up = ((t & k) == 0);
                    float a = ss[t], c = ss[x];
                    if (up ? (a < c) : (a > c)) {
                        ss[t] = c; ss[x] = a;
                        int ia = ii[t]; ii[t] = ii[x]; ii[x] = ia;
                    }
                }
            }
            __syncthreads();
        }
    }

    if (tid < K_TOP) {
        float s = ss[tid]; int idx = ii[tid];
        float prob = 0.0f; int c = 0; float4 bx = make_float4(0, 0, 0, 0);
        if (idx >= 0) {
            prob = 1.0f / (1.0f + __expf(-s));        // sigmoid only on survivors
            c    = cls_ws[b * N_ANC + idx];
            bx   = boxes_ws[b * N_ANC + idx];
        }
        top_s[b * K_TOP + tid] = prob;
        top_c[b * K_TOP + tid] = c;
        top_b[b * K_TOP + tid] = bx;
    }
}

// ---------------- kernel 6: greedy NMS (class-offset boxes) + final write ----------------
__global__ __launch_bounds__(1024)
void nms_kernel(const float* __restrict__ top_s, const int* __restrict__ top_c,
                const float4* __restrict__ top_b, float* __restrict__ out) {
    __shared__ float px1[K_TOP], py1[K_TOP], px2[K_TOP], py2[K_TOP], area[K_TOP];
    __shared__ int   keep[K_TOP];
    __shared__ float red[1024];
    __shared__ float maxc_s;

    const int b = blockIdx.x, j = threadIdx.x;

    float s = 0.0f; int c = 0; float4 bx = make_float4(0, 0, 0, 0); bool valid = false;
    if (j < K_TOP) {
        s  = top_s[b * K_TOP + j];
        c  = top_c[b * K_TOP + j];
        bx = top_b[b * K_TOP + j];
        valid = (s >= 0.05f);
    }

    // maxc = max coord over valid boxes (0 if none)
    float lm = -3.0e38f;
    if (valid) lm = fmaxf(fmaxf(bx.x, bx.y), fmaxf(bx.z, bx.w));
    red[j] = lm;
    __syncthreads();
    for (int st = 512; st > 0; st >>= 1) {
        if (j < st) red[j] = fmaxf(red[j], red[j + st]);
        __syncthreads();
    }
    if (j == 0) { float m = red[0]; maxc_s = (m > -1.0e37f) ? m : 0.0f; }
    __syncthreads();

    if (j < K_TOP) {
        float off = (float)c * (maxc_s + 1.0f);
        float x1 = bx.x + off, y1 = bx.y + off, x2 = bx.z + off, y2 = bx.w + off;
        px1[j] = x1; py1[j] = y1; px2[j] = x2; py2[j] = y2;
        area[j] = (x2 - x1 + 1.0f) * (y2 - y1 + 1.0f);
        keep[j] = valid ? 1 : 0;
    }
    __syncthreads();

    for (int i = 0; i < K_TOP - 1; ++i) {
        int ki = keep[i];
        if (ki && j > i && j < K_TOP && keep[j]) {
            float xx1 = fmaxf(px1[i], px1[j]);
            float yy1 = fmaxf(py1[i], py1[j]);
            float xx2 = fminf(px2[i], px2[j]);
            float yy2 = fminf(py2[i], py2[j]);
            float inter = fmaxf(0.0f, xx2 - xx1) * fmaxf(0.0f, yy2 - yy1);
            float iou = inter / (area[i] + area[j] - inter + 1e-16f);
            if (iou > 0.5f) keep[j] = 0;
        }
        __syncthreads();
    }

    if (j < K_TOP) {
        int k = keep[j];
        float* out_s  = out;                       // [8,1000]
        float* out_c  = out + B_IMG * K_TOP;       // [8,1000]
        float* out_bx = out + 2 * B_IMG * K_TOP;   // [8,1000,4]
        float* out_k  = out + 6 * B_IMG * K_TOP;   // [8,1000]
        int o = b * K_TOP + j;
        out_s[o] = k ? s : 0.0f;
        out_c[o] = k ? (float)c : 0.0f;
        float4 ob = k ? bx : make_float4(0, 0, 0, 0);
        out_bx[o * 4 + 0] = ob.x; out_bx[o * 4 + 1] = ob.y;
        out_bx[o * 4 + 2] = ob.z; out_bx[o * 4 + 3] = ob.w;
        out_k[o] = k ? 1.0f : 0.0f;
    }
}

// ---------------- host launcher ----------------
extern "C" void kernel_launch(void* const* d_in, const int* in_sizes, int n_in,
                              void* d_out, int out_size, void* d_ws, size_t ws_size,
                              hipStream_t stream) {
    (void)in_sizes; (void)n_in; (void)out_size; (void)ws_size;
    const float*  cls_logits = (const float*)d_in[0];   // [8,120000,80]
    const float4* reg_preds  = (const float4*)d_in[1];  // [8,120000,4]
    const float4* anchors    = (const float4*)d_in[2];  // [120000,4]
    float* out = (float*)d_out;

    // workspace carve-up (all chunks 16B aligned)
    char* p = (char*)d_ws;
    float*  scores_ws  = (float*)p;            p += (size_t)BN * 4;                 // 3.84 MB
    int*    cls_ws     = (int*)p;              p += (size_t)BN * 4;                 // 3.84 MB
    float4* boxes_ws   = (float4*)p;           p += (size_t)BN * 16;                // 15.36 MB
    int*    hist       = (int*)p;              p += (size_t)B_IMG * HBINS * 4;      // 64 KB
    int*    cutoff     = (int*)p;              p += 32;
    int*    cand_count = (int*)p;              p += 32;
    float*  cand_s     = (float*)p;            p += (size_t)B_IMG * CANDMAX * 4;    // 128 KB
    int*    cand_i     = (int*)p;              p += (size_t)B_IMG * CANDMAX * 4;    // 128 KB
    float*  top_s      = (float*)p;            p += (size_t)B_IMG * K_TOP * 4;
    int*    top_c      = (int*)p;              p += (size_t)B_IMG * K_TOP * 4;
    float4* top_b      = (float4*)p;           p += (size_t)B_IMG * K_TOP * 16;

    zero_meta<<<(B_IMG * HBINS + 255) / 256, 256, 0, stream>>>(hist, cutoff, cand_count);

    score_decode<<<BN / ROWS_PER_BLK, ROWS_PER_BLK, 0, stream>>>(
        cls_logits, reg_preds, anchors, scores_ws, cls_ws, boxes_ws);

    hist_kernel<<<(BN + 255) / 256, 256, 0, stream>>>(scores_ws, hist);
    cutoff_kernel<<<B_IMG, 32, 0, stream>>>(hist, cutoff);
    compact_kernel<<<(BN + 255) / 256, 256, 0, stream>>>(
        scores_ws, cutoff, cand_count, cand_s, cand_i);

    topk_kernel<<<B_IMG, 1024, 0, stream>>>(
        cand_s, cand_i, cand_count, cls_ws, boxes_ws, top_s, top_c, top_b);

    nms_kernel<<<B_IMG, 1024, 0, stream>>>(top_s, top_c, top_b, out);
}